// MultiviewTemporalSpatialFeatureAggregator_867583394043
// MI455X (gfx1250) — compile-verified
//
#include <hip/hip_runtime.h>
#include <hip/hip_bf16.h>

// ---------------------------------------------------------------------------
// Problem constants (from reference): B=1, N=128, D=C=256, T=4, Ncam=6, P=17,
// L=3, WC = 6*17*3 = 306. Levels: (64,160), (32,80), (16,40).
// ---------------------------------------------------------------------------
#define NQ   128
#define DC   256
#define TT   4
#define NCAM 6
#define PP   17
#define LL   3
#define WC   306
#define NSAMP (TT*NCAM*PP*LL)   // 1224 samples per query

typedef float v2f __attribute__((ext_vector_type(2)));
typedef float v8f __attribute__((ext_vector_type(8)));

// ---------------------------------------------------------------------------
// Kernel 1: logits = cq @ Ww, epilogue sigmoid(+bias) -> w0[NQ][WC]
// Full-f32 WMMA (V_WMMA_F32_16X16X4_F32), one wave per 16x16 output tile.
// M tiles = 128/16 = 8, N tiles = ceil(306/16) = 20 -> 160 waves.
// A frag (16x4 f32): lanes 0-15 row=lane, {K,K+1}; lanes 16-31 row=lane-16,
// {K+2,K+3}. B frag (4x16): row-striped, VGPR0={K,K+2}, VGPR1={K+1,K+3},
// col = lane%16. C/D: VGPR v -> row v + 8*(lane/16), col = lane%16.
//
// Boundary handling: the last N-tile covers cols 306..319. Instead of
// predicating the B loads (which injects saveexec/branch into the hot loop),
// we clamp the load column to WC-1; the resulting accumulator lanes are
// garbage but are never stored (store epilogue keeps the col<WC guard).
// This keeps the K-loop branch-free: 2 loads + 1 WMMA per step, EXEC
// untouched (WMMA requires EXEC all-1s).
// ---------------------------------------------------------------------------
__global__ void wmma_proj_kernel(const float* __restrict__ cq,   // [NQ][DC]
                                 const float* __restrict__ Ww,   // [DC][WC]
                                 const float* __restrict__ bw,   // [WC]
                                 float* __restrict__ w0)         // [NQ][WC]
{
    int tile = blockIdx.x;        // 0..159
    int tm = tile / 20;           // 0..7
    int tn = tile % 20;           // 0..19
    int lane = threadIdx.x & 31;
    int half = lane >> 4;         // 0 or 1
    int l16  = lane & 15;
    int arow = tm * 16 + l16;     // A row this lane feeds
    int col  = tn * 16 + l16;     // B/C/D column this lane holds
    int colc = (col < WC) ? col : (WC - 1);   // clamped load column

    const float* ap = cq + arow * DC;         // contiguous K walk (b64 loads)
    const float* bp = Ww + colc;              // strided by WC per K row

    v8f acc = {};
    for (int k = 0; k < DC; k += 4) {
        int ka = k + half * 2;
        v2f a, b;
        a.x = ap[ka];
        a.y = ap[ka + 1];
        b.x = bp[ka * WC];
        b.y = bp[(ka + 1) * WC];
        acc = __builtin_amdgcn_wmma_f32_16x16x4_f32(
            /*neg_a=*/false, a, /*neg_b=*/false, b,
            /*c_mod=*/(short)0, acc, /*reuse_a=*/false, /*reuse_b=*/false);
    }

    if (col < WC) {
        float bias = bw[col];
        #pragma unroll
        for (int v = 0; v < 8; ++v) {
            int r = tm * 16 + v + 8 * half;
            float x = acc[v] + bias;
            w0[r * WC + col] = 1.0f / (1.0f + __expf(-x));
        }
    }
}

// ---------------------------------------------------------------------------
// Kernel 2: layout transpose per level: [S=24][C=256][HW] -> [S][HW][C]
// so channel gathers in the aggregator are contiguous (1 cacheline / wave).
// LDS 32x33 tile; reads coalesced along HW, writes coalesced along C.
// ---------------------------------------------------------------------------
__global__ void transpose_kernel(const float* __restrict__ in,
                                 float* __restrict__ out, int HW)
{
    __shared__ float tile[32][33];
    int s   = blockIdx.z;                 // cam*4 + bt
    int hw0 = blockIdx.x * 32;
    int c0  = blockIdx.y * 32;
    const float* ip = in  + (size_t)s * DC * HW;
    float*       op = out + (size_t)s * HW * DC;
    int lx = threadIdx.x;                 // 0..31
    int ly = threadIdx.y;                 // 0..7
    #pragma unroll
    for (int i = 0; i < 32; i += 8)
        tile[ly + i][lx] = ip[(size_t)(c0 + ly + i) * HW + hw0 + lx];
    __syncthreads();
    #pragma unroll
    for (int i = 0; i < 32; i += 8)
        op[(size_t)(hw0 + ly + i) * DC + c0 + lx] = tile[lx][ly + i];
}

// ---------------------------------------------------------------------------
// Kernel 3: aggregation. One workgroup per query n (128 WGs), 256 threads
// (= channels, 8 wave32s).
// Phase 1: threads cooperatively build 1224 sample records in LDS:
//   4 clamped corner offsets, 4 weights premultiplied by
//   sigmoid-weight * tw[t] * validity, plus level id; accumulate den.
// Phase 2: each thread streams its channel through all records (coalesced
//   when features are transposed), then writes concat(cq, num/den).
// ---------------------------------------------------------------------------
struct SampleRec {
    int   off[4];
    float wt[4];
    int   lvl;
};

__global__ void aggregate_kernel(const float* __restrict__ cq,     // [NQ][DC]
                                 const float* __restrict__ pixels, // [TT][NQ][NCAM][PP][2]
                                 const float* __restrict__ ego,    // [TT]
                                 const float* __restrict__ wsig,   // [NQ][WC]
                                 const float* __restrict__ f0,
                                 const float* __restrict__ f1,
                                 const float* __restrict__ f2,
                                 int transposed,
                                 float* __restrict__ out_ncq)      // [NQ][2*DC]
{
    __shared__ SampleRec rec[NSAMP];
    __shared__ float red[256];
    __shared__ float s_den;

    int n   = blockIdx.x;
    int tid = threadIdx.x;      // channel index in phase 2
    float egoT = ego[TT - 1];
    float denp = 0.0f;

    for (int s = tid; s < NSAMP; s += 256) {
        int l   = s % LL;
        int r   = s / LL;
        int p   = r % PP;  r /= PP;
        int cam = r % NCAM; r /= NCAM;
        int t   = r;                    // 0..3 (B==1 so bt==t)
        int bt  = t;

        int H = (l == 0) ? 64  : (l == 1) ? 32 : 16;
        int W = (l == 0) ? 160 : (l == 1) ? 80 : 40;

        float td = ego[t] - egoT;
        float tw = __expf(-td * td * 6.0f);          // exp(-td^2/0.5*3)

        int pix = (((bt * NQ + n) * NCAM + cam) * PP + p) * 2;
        float px0 = pixels[pix + 0];
        float px1 = pixels[pix + 1];

        float wsv = wsig[n * WC + (cam * PP + p) * LL + l] * tw;
        if (px0 < 0.0f) wsv = 0.0f;
        denp += wsv;

        float gx = px0 * 2.0f - 1.0f;
        float gy = px1 * 2.0f - 1.0f;
        float ix = (gx + 1.0f) * 0.5f * (float)(W - 1);
        float iy = (gy + 1.0f) * 0.5f * (float)(H - 1);
        float x0f = floorf(ix), y0f = floorf(iy);
        float wx1 = ix - x0f, wy1 = iy - y0f;
        float wx0 = 1.0f - wx1, wy0 = 1.0f - wy1;

        // slab base for (cam, bt) in element units
        int HWl  = H * W;
        int slab = transposed ? (cam * TT + bt) * HWl * DC
                              : (cam * TT + bt) * DC * HWl;

        SampleRec sr;
        sr.lvl = l;
        float cw[4]  = { wx0 * wy0, wx1 * wy0, wx0 * wy1, wx1 * wy1 };
        float cx[4]  = { x0f, x0f + 1.0f, x0f, x0f + 1.0f };
        float cyv[4] = { y0f, y0f, y0f + 1.0f, y0f + 1.0f };
        #pragma unroll
        for (int j = 0; j < 4; ++j) {
            float xf = cx[j], yf = cyv[j];
            bool valid = (xf >= 0.0f) && (xf <= (float)(W - 1)) &&
                         (yf >= 0.0f) && (yf <= (float)(H - 1));
            int xc = (int)fminf(fmaxf(xf, 0.0f), (float)(W - 1));
            int yc = (int)fminf(fmaxf(yf, 0.0f), (float)(H - 1));
            int pos = yc * W + xc;
            sr.off[j] = transposed ? (slab + pos * DC) : (slab + pos);
            sr.wt[j]  = valid ? (wsv * cw[j]) : 0.0f;
        }
        rec[s] = sr;
    }

    // reduce den across the workgroup
    red[tid] = denp;
    __syncthreads();
    for (int st = 128; st > 0; st >>= 1) {
        if (tid < st) red[tid] += red[tid + st];
        __syncthreads();
    }
    if (tid == 0) s_den = fmaxf(red[0], 1e-6f);
    __syncthreads();
    float den = s_den;

    // per-level channel stride
    int cs0 = transposed ? 1 : 64 * 160;
    int cs1 = transposed ? 1 : 32 * 80;
    int cs2 = transposed ? 1 : 16 * 40;

    int c = tid;
    float num = 0.0f;
    #pragma unroll 2
    for (int s = 0; s < NSAMP; ++s) {
        SampleRec r = rec[s];            // broadcast LDS read (all lanes same)
        const float* bp;
        int cst;
        if (r.lvl == 0)      { bp = f0; cst = cs0; }
        else if (r.lvl == 1) { bp = f1; cst = cs1; }
        else                 { bp = f2; cst = cs2; }
        int cc = c * cst;
        num += r.wt[0] * bp[r.off[0] + cc]
             + r.wt[1] * bp[r.off[1] + cc]
             + r.wt[2] * bp[r.off[2] + cc]
             + r.wt[3] * bp[r.off[3] + cc];
    }

    out_ncq[n * (2 * DC) + c]      = cq[n * DC + c];
    out_ncq[n * (2 * DC) + DC + c] = num / den;
}

// ---------------------------------------------------------------------------
// Launch
// ---------------------------------------------------------------------------
extern "C" void kernel_launch(void* const* d_in, const int* in_sizes, int n_in,
                              void* d_out, int out_size, void* d_ws, size_t ws_size,
                              hipStream_t stream) {
    (void)in_sizes; (void)n_in; (void)out_size;
    const float* cq     = (const float*)d_in[0];
    const float* pixels = (const float*)d_in[1];
    const float* ego    = (const float*)d_in[2];
    const float* Ww     = (const float*)d_in[3];
    const float* bw     = (const float*)d_in[4];
    const float* f0     = (const float*)d_in[5];
    const float* f1     = (const float*)d_in[6];
    const float* f2     = (const float*)d_in[7];

    float* out = (float*)d_out;
    const size_t PIXN = (size_t)TT * NQ * NCAM * PP * 2;   // 104448

    // output[0]: pixels passthrough (async d2d copy is graph-capture safe)
    hipMemcpyAsync(out, pixels, PIXN * sizeof(float),
                   hipMemcpyDeviceToDevice, stream);
    float* ncq = out + PIXN;

    // ws layout: [0, 1MB) -> w0 sigmoid weights; [1MB, ...) -> transposed feats
    float* w0 = (float*)d_ws;
    wmma_proj_kernel<<<160, 32, 0, stream>>>(cq, Ww, bw, w0);

    const size_t T0E = (size_t)NCAM * TT * 64 * 160 * DC;  // 62,914,560
    const size_t T1E = (size_t)NCAM * TT * 32 * 80  * DC;  // 15,728,640
    const size_t T2E = (size_t)NCAM * TT * 16 * 40  * DC;  //  3,932,160
    const size_t TOFF = (size_t)1 << 20;                   // 1 MB
    int transposed = 0;
    const float *g0 = f0, *g1 = f1, *g2 = f2;
    if (ws_size >= TOFF + (T0E + T1E + T2E) * sizeof(float)) {
        transposed = 1;
        float* t0 = (float*)((char*)d_ws + TOFF);
        float* t1 = t0 + T0E;
        float* t2 = t1 + T1E;
        transpose_kernel<<<dim3(10240 / 32, DC / 32, NCAM * TT), dim3(32, 8), 0, stream>>>(f0, t0, 10240);
        transpose_kernel<<<dim3(2560  / 32, DC / 32, NCAM * TT), dim3(32, 8), 0, stream>>>(f1, t1, 2560);
        transpose_kernel<<<dim3(640   / 32, DC / 32, NCAM * TT), dim3(32, 8), 0, stream>>>(f2, t2, 640);
        g0 = t0; g1 = t1; g2 = t2;
    }

    aggregate_kernel<<<NQ, 256, 0, stream>>>(cq, pixels, ego, w0,
                                             g0, g1, g2, transposed, ncq);
}